// Ewald_4741643895682
// MI455X (gfx1250) — compile-verified
//
#include <hip/hip_runtime.h>
#include <math.h>

typedef __attribute__((ext_vector_type(2))) float v2f;
typedef __attribute__((ext_vector_type(8))) float v8f;

#define EWALD_COEF 14.399645478425668f
#define EWALD_EPS  1e-8f
#define PI_F 3.14159265358979323846f
#define MAXN  1024
#define MAXT  (MAXN/16)

struct CellInfo {
  float c[9];   // cell rows (row-major 3x3)
  float r[9];   // recip rows: 2*pi*inv(cell)^T, row-major
  float vol, eta, cutR, cutK, inv_sqrt2eta, eta2;
};

__device__ inline void compute_cell(const float* __restrict__ cell, int N, CellInfo& ci) {
#pragma unroll
  for (int i = 0; i < 9; i++) ci.c[i] = cell[i];
  float a00=ci.c[0],a01=ci.c[1],a02=ci.c[2];
  float a10=ci.c[3],a11=ci.c[4],a12=ci.c[5];
  float a20=ci.c[6],a21=ci.c[7],a22=ci.c[8];
  float cof00 =  (a11*a22 - a12*a21);
  float cof01 = -(a10*a22 - a12*a20);
  float cof02 =  (a10*a21 - a11*a20);
  float cof10 = -(a01*a22 - a02*a21);
  float cof11 =  (a00*a22 - a02*a20);
  float cof12 = -(a00*a21 - a01*a20);
  float cof20 =  (a01*a12 - a02*a11);
  float cof21 = -(a00*a12 - a02*a10);
  float cof22 =  (a00*a11 - a01*a10);
  float det = a00*cof00 + a01*cof01 + a02*cof02;
  float vol = fabsf(det);
  float tp = (2.0f*PI_F)/det;             // recip[i][j] = 2*pi*cof[i][j]/det
  ci.r[0]=cof00*tp; ci.r[1]=cof01*tp; ci.r[2]=cof02*tp;
  ci.r[3]=cof10*tp; ci.r[4]=cof11*tp; ci.r[5]=cof12*tp;
  ci.r[6]=cof20*tp; ci.r[7]=cof21*tp; ci.r[8]=cof22*tp;
  ci.vol = vol;
  float eta = powf(vol*vol/(float)N, 1.0f/6.0f) / sqrtf(2.0f*PI_F);
  ci.eta = eta;
  float s2l = sqrtf(-2.0f*logf(1e-8f));   // SQRT_2LOG
  ci.cutR = s2l*eta;
  ci.cutK = s2l/eta;
  ci.inv_sqrt2eta = 1.0f/(sqrtf(2.0f)*eta);
  ci.eta2 = eta*eta;
}

// Branchless per-shift accumulation, mirrors reference's where() masking.
__device__ __forceinline__ float shift_term(float dx, float dy, float dz,
                                            float sx, float sy, float sz,
                                            float cutR, float inv_s2eta, float inv_s2gam) {
  float ex = dx+sx, ey = dy+sy, ez = dz+sz;
  float dist = sqrtf(ex*ex + ey*ey + ez*ez);
  bool ok = (dist > EWALD_EPS) && (dist < cutR);
  float safe = ok ? dist : 1.0f;
  float v = (erfcf(safe*inv_s2eta) - erfcf(safe*inv_s2gam)) / safe;
  return ok ? v : 0.0f;
}

template<int NS>
__device__ __forceinline__ float real_pair_sum_fixed(float dx, float dy, float dz,
                                                     const CellInfo& ci, float inv_s2gam) {
  float sum = 0.0f;
#pragma unroll
  for (int i0=-NS;i0<=NS;i0++)
#pragma unroll
    for (int i1=-NS;i1<=NS;i1++)
#pragma unroll
      for (int i2=-NS;i2<=NS;i2++) {
        float sx = i0*ci.c[0]+i1*ci.c[3]+i2*ci.c[6];
        float sy = i0*ci.c[1]+i1*ci.c[4]+i2*ci.c[7];
        float sz = i0*ci.c[2]+i1*ci.c[5]+i2*ci.c[8];
        sum += shift_term(dx,dy,dz,sx,sy,sz,ci.cutR,ci.inv_sqrt2eta,inv_s2gam);
      }
  return sum;
}

__device__ __forceinline__ float real_pair_sum_dyn(float dx, float dy, float dz,
                                                   const CellInfo& ci, float inv_s2gam, int ns) {
  float sum = 0.0f;
  for (int i0=-ns;i0<=ns;i0++)
    for (int i1=-ns;i1<=ns;i1++)
      for (int i2=-ns;i2<=ns;i2++) {
        float sx = i0*ci.c[0]+i1*ci.c[3]+i2*ci.c[6];
        float sy = i0*ci.c[1]+i1*ci.c[4]+i2*ci.c[7];
        float sz = i0*ci.c[2]+i1*ci.c[5]+i2*ci.c[8];
        sum += shift_term(dx,dy,dz,sx,sy,sz,ci.cutR,ci.inv_sqrt2eta,inv_s2gam);
      }
  return sum;
}

// ------------- Real-space kernel: upper triangle, weight-2 off-diagonal -------------
__global__ void __launch_bounds__(256)
ewald_real_kernel(const float* __restrict__ pos, const float* __restrict__ cell,
                  const float* __restrict__ charges, const float* __restrict__ sigma_table,
                  const int* __restrict__ species, const int* __restrict__ nshift_real,
                  int N, float* __restrict__ partials)
{
  __shared__ float red[256];
  CellInfo ci; compute_cell(cell, N, ci);
  int ns = nshift_real[0];
  float acc = 0.0f;
  long total = (long)N*(N+1)/2;
  long stride = (long)gridDim.x*blockDim.x;
  for (long p = blockIdx.x*blockDim.x + threadIdx.x; p < total; p += stride) {
    // decode triangle index p -> (a,b) with a <= b
    double tn = 2.0*(double)N + 1.0;
    int a = (int)((tn - sqrt(tn*tn - 8.0*(double)p)) * 0.5);
    if (a < 0) a = 0;
    if (a > N-1) a = N-1;
    long off = (long)a*N - ((long)a*(a-1))/2;
    while (p < off) { --a; off = (long)a*N - ((long)a*(a-1))/2; }
    while (a < N-1) {
      long off2 = (long)(a+1)*N - ((long)(a+1)*a)/2;
      if (p >= off2) { ++a; off = off2; } else break;
    }
    int b = a + (int)(p - off);

    float dx = pos[3*b+0]-pos[3*a+0];
    float dy = pos[3*b+1]-pos[3*a+1];
    float dz = pos[3*b+2]-pos[3*a+2];
    float sa = sigma_table[species[a]];
    float sb = sigma_table[species[b]];
    float gam = sqrtf(sa*sa+sb*sb);
    float inv_s2gam = 1.0f/(sqrtf(2.0f)*gam);
    float sum = (ns == 1) ? real_pair_sum_fixed<1>(dx,dy,dz,ci,inv_s2gam)
                          : real_pair_sum_dyn(dx,dy,dz,ci,inv_s2gam,ns);
    float w = (a == b) ? 1.0f : 2.0f;
    acc += w * EWALD_COEF * charges[a]*charges[b] * sum;
  }
  red[threadIdx.x] = acc;
  __syncthreads();
  for (int s = 128; s > 0; s >>= 1) {
    if (threadIdx.x < s) red[threadIdx.x] += red[threadIdx.x + s];
    __syncthreads();
  }
  if (threadIdx.x == 0) partials[blockIdx.x] = red[0];
}

// ---------- Reciprocal-space kernel: theta tiles via V_WMMA_F32_16X16X4_F32 ----------
// Per wave: tile of 16 k-vectors (A, 16x4 with padded K=4) x tiles of 16 atoms (B, 4x16,
// precomputed in LDS as packed float2). D = theta (16x16 f32). Structure factors are
// accumulated per lane, butterfly-reduced over 16-lane groups; per-row weights w(k) are
// computed once per lane at A-build time and routed with a single width-16 shuffle.
__global__ void __launch_bounds__(256)
ewald_recip_kernel(const float* __restrict__ pos, const float* __restrict__ cell,
                   const float* __restrict__ charges, const int* __restrict__ nshift_recip,
                   int N, float* __restrict__ partials)
{
  __shared__ v2f   sB[MAXT*32];   // precomputed B operand per (atom-tile, lane)
  __shared__ float sqpad[MAXN];   // zero-padded charges
  __shared__ float red[256];

  CellInfo ci; compute_cell(cell, N, ci);
  int nk = nshift_recip[0];
  int dim = 2*nk+1;
  int K = dim*dim*dim;
  int ktiles = (K+15)>>4;
  int atiles = (N+15)>>4;

  // Precompute B tiles: B[K=0..3][N=n] = (px,py,pz,0) of atom n, laid out per WMMA B spec.
  for (int idx = threadIdx.x; idx < atiles*32; idx += blockDim.x) {
    int at = idx >> 5, l = idx & 31, mm = l & 15, hh = l >> 4;
    int aidx = at*16 + mm;
    int ac = aidx < N ? aidx : N-1;
    float px = pos[3*ac+0], py = pos[3*ac+1], pz = pos[3*ac+2];
    v2f B;
    B.x = hh ? pz   : px;   // VGPR0: row K=0 | K=2
    B.y = hh ? 0.0f : py;   // VGPR1: row K=1 | K=3
    sB[idx] = B;
  }
  for (int i = threadIdx.x; i < atiles*16; i += blockDim.x)
    sqpad[i] = (i < N) ? charges[i] : 0.0f;
  __syncthreads();

  int lane = threadIdx.x & 31;
  int m    = lane & 15;
  int hi   = lane >> 4;
  int wave  = (int)((blockIdx.x*blockDim.x + threadIdx.x) >> 5);
  int nwave = (int)((gridDim.x*blockDim.x) >> 5);
  float prefac = EWALD_COEF * 4.0f * PI_F / ci.vol;
  float acc = 0.0f;

  for (int kt = wave; kt < ktiles; kt += nwave) {
    // Build A (16x4 f32): lane m supplies row M=m. VGPR0: K=0|K=2, VGPR1: K=1|K=3.
    int kidxA = kt*16 + m;
    int ic = kidxA < K ? kidxA : K-1;
    int iz = ic % dim; int tq = ic / dim; int iy = tq % dim; int ix = tq / dim;
    float mx = (float)(ix - nk), my = (float)(iy - nk), mz = (float)(iz - nk);
    float kx = mx*ci.r[0] + my*ci.r[3] + mz*ci.r[6];
    float ky = mx*ci.r[1] + my*ci.r[4] + mz*ci.r[7];
    float kz = mx*ci.r[2] + my*ci.r[5] + mz*ci.r[8];
    float vA = (kidxA < K) ? 1.0f : 0.0f;     // branchless pad, EXEC stays uniform
    kx *= vA; ky *= vA; kz *= vA;
    // Per-row reciprocal weight w(k), computed once per lane (row m of this tile).
    float k2A = kx*kx + ky*ky + kz*kz;
    float klA = sqrtf(k2A);
    bool kok = (klA > EWALD_EPS) && (klA < ci.cutK);
    float wA = kok ? prefac*__expf(-0.5f*ci.eta2*k2A)/k2A : 0.0f;

    v2f A;
    A.x = hi ? kz   : kx;   // K=0 / K=2
    A.y = hi ? 0.0f : ky;   // K=1 / K=3

    float accC[8], accS[8];
#pragma unroll
    for (int r=0;r<8;r++){accC[r]=0.0f;accS[r]=0.0f;}

#pragma unroll 2
    for (int at = 0; at < atiles; ++at) {
      v2f B = sB[at*32 + lane];
      float qv = sqpad[at*16 + m];
      v8f cz = {0.f,0.f,0.f,0.f,0.f,0.f,0.f,0.f};
      v8f th = __builtin_amdgcn_wmma_f32_16x16x4_f32(
          /*neg_a=*/false, A, /*neg_b=*/false, B,
          /*c_mod=*/(short)0, cz, /*reuse_a=*/false, /*reuse_b=*/false);
#pragma unroll
      for (int r=0;r<8;r++) {
        float sv, cv;
        __sincosf(th[r], &sv, &cv);
        accC[r] += qv*cv;
        accS[r] += qv*sv;
      }
    }

    // Reduce structure factors over the 16 atom-columns of each half-group,
    // then lane m==r accumulates w(row r+8*hi) * |S|^2.
#pragma unroll
    for (int r=0;r<8;r++) {
      float c = accC[r], s = accS[r];
      c += __shfl_xor(c, 8, 16);  s += __shfl_xor(s, 8, 16);
      c += __shfl_xor(c, 4, 16);  s += __shfl_xor(s, 4, 16);
      c += __shfl_xor(c, 2, 16);  s += __shfl_xor(s, 2, 16);
      c += __shfl_xor(c, 1, 16);  s += __shfl_xor(s, 1, 16);
      float wr = __shfl(wA, r + 8*hi, 16);   // row M = r + 8*half's weight
      acc += (m == r) ? wr*(c*c + s*s) : 0.0f;
    }
  }

  red[threadIdx.x] = acc;
  __syncthreads();
  for (int s2 = 128; s2 > 0; s2 >>= 1) {
    if (threadIdx.x < s2) red[threadIdx.x] += red[threadIdx.x + s2];
    __syncthreads();
  }
  if (threadIdx.x == 0) partials[blockIdx.x] = red[0];
}

// ---------------- Finalize: sum partials + self energy, deterministic ----------------
__global__ void __launch_bounds__(256)
ewald_finalize_kernel(const float* __restrict__ partials, int nPart,
                      const float* __restrict__ cell, const float* __restrict__ charges,
                      const float* __restrict__ sigma_table, const int* __restrict__ species,
                      int N, float* __restrict__ out)
{
  __shared__ float red[256];
  CellInfo ci; compute_cell(cell, N, ci);
  float acc = 0.0f;
  for (int i = threadIdx.x; i < nPart; i += blockDim.x) acc += partials[i];
  float selfc = -sqrtf(2.0f/PI_F)/ci.eta;
  float invsqrtpi = 1.0f/sqrtf(PI_F);
  for (int i = threadIdx.x; i < N; i += blockDim.x) {
    float qi = charges[i];
    float sg = sigma_table[species[i]];
    acc += EWALD_COEF * qi*qi * (selfc + invsqrtpi/sg);
  }
  red[threadIdx.x] = acc;
  __syncthreads();
  for (int s = 128; s > 0; s >>= 1) {
    if (threadIdx.x < s) red[threadIdx.x] += red[threadIdx.x + s];
    __syncthreads();
  }
  if (threadIdx.x == 0) out[0] = 0.5f * red[0];
}

extern "C" void kernel_launch(void* const* d_in, const int* in_sizes, int n_in,
                              void* d_out, int out_size, void* d_ws, size_t ws_size,
                              hipStream_t stream)
{
  const float* pos         = (const float*)d_in[0];
  const float* cell        = (const float*)d_in[1];
  const float* charges     = (const float*)d_in[2];
  const float* sigma_table = (const float*)d_in[3];
  const int*   species     = (const int*)  d_in[4];
  const int*   nshift_real = (const int*)  d_in[5];
  const int*   nshift_recp = (const int*)  d_in[6];
  int N = in_sizes[0] / 3;
  float* out = (float*)d_out;
  float* ws  = (float*)d_ws;

  const int NBR = 256;   // real-space partial blocks (triangle pairs)
  const int NBK = 64;    // recip-space partial blocks
  float* pReal  = ws;         // ws[0 .. NBR)
  float* pRecip = ws + NBR;   // ws[NBR .. NBR+NBK)

  ewald_real_kernel<<<NBR, 256, 0, stream>>>(pos, cell, charges, sigma_table,
                                             species, nshift_real, N, pReal);
  ewald_recip_kernel<<<NBK, 256, 0, stream>>>(pos, cell, charges, nshift_recp, N, pRecip);
  ewald_finalize_kernel<<<1, 256, 0, stream>>>(ws, NBR + NBK, cell, charges,
                                               sigma_table, species, N, out);
}